// BoxFilter_19602230739840
// MI455X (gfx1250) — compile-verified
//
#include <hip/hip_runtime.h>

// 9x9 (radius-4) unnormalized box sum with zero padding == the reference's
// cumsum + diff_x/diff_y pipeline. Separable, LDS-tiled, CDNA5 async
// global->LDS fill (ASYNCcnt path), non-temporal output stores.

#define R       4
#define TH      128                  // output tile rows
#define TW      64                   // output tile cols
#define HALO_H  (TH + 2 * R)         // 136
#define HALO_W  (TW + 2 * R)         // 72
#define HW4     (HALO_W / 4)         // 18 float4 groups per halo row
#define NTHREADS 256

typedef __attribute__((ext_vector_type(4))) float f32x4;  // native vector (NT-store legal)

__device__ __forceinline__ void wait_async_zero() {
#if defined(__has_builtin)
#if __has_builtin(__builtin_amdgcn_s_wait_asynccnt)
  __builtin_amdgcn_s_wait_asynccnt(0);
#else
  asm volatile("s_wait_asynccnt 0x0" ::: "memory");
#endif
#else
  asm volatile("s_wait_asynccnt 0x0" ::: "memory");
#endif
}

__global__ __launch_bounds__(NTHREADS) void box9_async_kernel(
    const float* __restrict__ in, float* __restrict__ out) {
  // Dynamic LDS: 136*72 halo tile + 136*64 horizontal sums = 73,984 bytes.
  extern __shared__ __align__(16) float smem[];
  float* tileA = smem;                        // [HALO_H][HALO_W]
  float* tileB = smem + HALO_H * HALO_W;      // [HALO_H][TW] (16B-aligned: 39168%16==0)

  const int tid = threadIdx.x;
  const int h0  = blockIdx.y * TH;
  const int w0  = blockIdx.x * TW;
  const size_t planeOff = (size_t)blockIdx.z * (512 * 512);
  const float* __restrict__ src = in + planeOff;

  // ---- Stage 1: zero-padded halo fill via async global->LDS (float4 granular).
  // w0 is a multiple of 64, so w0-4 is 16B aligned and every halo float4 is
  // either fully in-range (async load, bypasses VGPRs) or fully OOB (zero).
  for (int g = tid; g < HALO_H * HW4; g += NTHREADS) {
    const int lr = g / HW4;                 // halo row 0..135
    const int gc = g - lr * HW4;            // float4 col 0..17
    const int gh = h0 + lr - R;             // global row
    const int gw = w0 + gc * 4 - R;         // global col (multiple of 4)
    const unsigned ldsAddr = (unsigned)(size_t)(&tileA[g * 4]);
    const bool valid = (gh >= 0) & (gh < 512) & (gw >= 0) & (gw < 512);
    if (valid) {
      const unsigned gOff = (unsigned)((gh * 512 + gw) * 4);  // byte offset in plane
      asm volatile("global_load_async_to_lds_b128 %0, %1, %2"
                   :
                   : "v"(ldsAddr), "v"(gOff), "s"(src)
                   : "memory");
    } else {
      *reinterpret_cast<f32x4*>(&tileA[g * 4]) = (f32x4){0.f, 0.f, 0.f, 0.f};
    }
  }
  wait_async_zero();
  __syncthreads();

  // ---- Stage 2: horizontal 9-tap sliding sums: tileB[lr][c] = sum tileA[lr][c..c+8]
  for (int u = tid; u < HALO_H * (TW / 4); u += NTHREADS) {
    const int lr = u / (TW / 4);
    const int c4 = u - lr * (TW / 4);       // output float4 col 0..15
    const float* rowp = &tileA[lr * HALO_W + c4 * 4];
    const f32x4 a  = *reinterpret_cast<const f32x4*>(rowp);
    const f32x4 b  = *reinterpret_cast<const f32x4*>(rowp + 4);
    const f32x4 c8 = *reinterpret_cast<const f32x4*>(rowp + 8);
    const float s07 = ((a.x + a.y) + (a.z + a.w)) + ((b.x + b.y) + (b.z + b.w));
    f32x4 o;
    o.x = s07 + c8.x;
    o.y = o.x - a.x + c8.y;
    o.z = o.y - a.y + c8.z;
    o.w = o.z - a.z + c8.w;
    *reinterpret_cast<f32x4*>(&tileB[lr * TW + c4 * 4]) = o;
  }
  __syncthreads();

  // ---- Stage 3: vertical 9-tap sums + coalesced non-temporal float4 stores.
  float* __restrict__ dst = out + planeOff;
  for (int u = tid; u < TH * (TW / 4); u += NTHREADS) {
    const int oy = u / (TW / 4);
    const int x4 = u - oy * (TW / 4);
    const float* colp = &tileB[oy * TW + x4 * 4];
    f32x4 acc = *reinterpret_cast<const f32x4*>(colp);
#pragma unroll
    for (int k = 1; k < 9; ++k) {
      acc += *reinterpret_cast<const f32x4*>(colp + k * TW);
    }
    const int gh = h0 + oy;
    f32x4* outp = reinterpret_cast<f32x4*>(dst + gh * 512 + w0 + x4 * 4);
    __builtin_nontemporal_store(acc, outp);   // write-once output: keep L2 for halos
  }
}

extern "C" void kernel_launch(void* const* d_in, const int* in_sizes, int n_in,
                              void* d_out, int out_size, void* d_ws, size_t ws_size,
                              hipStream_t stream) {
  (void)n_in; (void)d_ws; (void)ws_size; (void)out_size;
  const float* image = (const float*)d_in[0];
  float* out = (float*)d_out;
  const int total  = in_sizes[0];               // 8*32*512*512
  const int planes = total / (512 * 512);       // 256
  const size_t smemBytes = (size_t)(HALO_H * HALO_W + HALO_H * TW) * sizeof(float);
  dim3 grid(512 / TW, 512 / TH, planes);
  box9_async_kernel<<<grid, NTHREADS, smemBytes, stream>>>(image, out);
}